// IMG_SPCNN_66778151518377
// MI455X (gfx1250) — compile-verified
//
#include <hip/hip_runtime.h>
#include <hip/hip_bf16.h>

typedef __attribute__((ext_vector_type(2))) float f32x2;
typedef __attribute__((ext_vector_type(8))) float f32x8;

// ---------------- problem dims (compile-time) ----------------
#define BATCH 16
#define H0 96
#define W0 320
#define C0 32
#define H1 48
#define W1 160
#define C1 32
#define H2 24
#define W2 80
#define C2 64
#define H3 12
#define W3 40
#define C3 128
#define FC_IN  (C3*H3*W3)   // 61440
#define FC_OUT 256
#define KSPLIT 64           // FC K-dimension split (61440/64 = 960)

#define NEG_BIG (-3.402823466e38f)

// ---------------- small utility kernels ----------------
__global__ void zero_f32(float* __restrict__ p, int n) {
    int i = blockIdx.x * blockDim.x + threadIdx.x;
    if (i < n) p[i] = 0.0f;
}

__global__ void scatter_mask(const int* __restrict__ coords, float* __restrict__ mask, int n) {
    int i = blockIdx.x * blockDim.x + threadIdx.x;
    if (i >= n) return;
    int b = coords[3*i], y = coords[3*i+1], x = coords[3*i+2];
    mask[((long)b * H0 + y) * W0 + x] = 1.0f;
}

// repack OIHW weights -> row-major (K = (ky*3+kx)*CIN + ci, N = cout), so B-matrix
// fragment loads are contiguous across lanes (N) and small-stride across K.
template<int CIN, int COUT>
__global__ void repack_weights(const float* __restrict__ w, float* __restrict__ wt) {
    int tid = blockIdx.x * blockDim.x + threadIdx.x;
    const int total = 9 * CIN * COUT;
    if (tid >= total) return;
    int n = tid % COUT;
    int k = tid / COUT;
    int plane = k / CIN, ci = k % CIN;
    int ky = plane / 3, kx = plane % 3;
    wt[tid] = w[((n * CIN + ci) * 3 + ky) * 3 + kx];
}

// ---------------- implicit-GEMM conv via fp32 WMMA ----------------
// One wave per 16(M spatial) x 16(N cout) tile.  D = A x B + C via
// V_WMMA_F32_16X16X4_F32.  A fragment: lane holds M=lane%16, K={2*hi, 2*hi+1}.
// B fragment: VGPR r holds K-row (k + r + 2*hi), N = lane%16.
template<int CIN, int COUT, int HIN, int WIN, int HOUT, int WOUT, bool USE_MASK, bool OUT_NCHW>
__global__ __launch_bounds__(32)
void conv3x3s2_wmma(const float* __restrict__ in,    // NHWC (BATCH,HIN,WIN,CIN)
                    const float* __restrict__ mask,  // (BATCH,HIN,WIN) float 0/1 (stage 1 only)
                    const float* __restrict__ wt,    // packed (9*CIN, COUT)
                    const float* __restrict__ bias,  // (COUT)
                    float* __restrict__ out)         // NHWC (...,COUT) or NCHW
{
    const int lane = threadIdx.x;
    const int lm = lane & 15;
    const int hi = lane >> 4;
    const int mBase = blockIdx.x * 16;
    const int nBase = blockIdx.y * 16;

    // A-row (spatial) decode for this lane
    const int m    = mBase + lm;
    const int bIdx = m / (HOUT * WOUT);
    const int rem  = m % (HOUT * WOUT);
    const int oy   = rem / WOUT;
    const int ox   = rem % WOUT;
    // B-col for this lane
    const int n = nBase + lm;

    f32x8 acc = {};
    #pragma unroll
    for (int ky = 0; ky < 3; ++ky) {
        const int iy = oy * 2 - 1 + ky;
        const bool rowok = (iy >= 0) && (iy < HIN);
        #pragma unroll
        for (int kx = 0; kx < 3; ++kx) {
            const int ix = ox * 2 - 1 + kx;
            const bool ok = rowok && (ix >= 0) && (ix < WIN);
            const long pix = ((long)bIdx * HIN + iy) * WIN + ix;
            const float* ap = in + pix * CIN + 2 * hi;
            float mval = 1.0f;
            if (USE_MASK) mval = ok ? mask[pix] : 0.0f;
            const int kbase = (ky * 3 + kx) * CIN;
            for (int ci = 0; ci < CIN; ci += 4) {
                f32x2 a = {0.0f, 0.0f};
                if (ok) {
                    a = *(const f32x2*)(ap + ci);   // contiguous channels -> b64 load
                    if (USE_MASK) a *= mval;        // fused sparse mask (no dense tensor)
                }
                const int k = kbase + ci + 2 * hi;
                const float* bp = wt + (long)k * COUT + n;
                f32x2 bf;
                bf.x = bp[0];
                bf.y = bp[COUT];
                acc = __builtin_amdgcn_wmma_f32_16x16x4_f32(
                          false, a, false, bf, (short)0, acc, false, false);
            }
        }
    }

    // epilogue: C/D layout -> M = r + 8*hi, N = lane%16 ; add bias
    const int col = nBase + lm;
    const float bb = bias[col];
    #pragma unroll
    for (int r = 0; r < 8; ++r) {
        const int row = mBase + r + 8 * hi;
        const float v = acc[r] + bb;
        if (OUT_NCHW) {
            const int b2 = row / (HOUT * WOUT);
            const int s  = row % (HOUT * WOUT);
            out[((long)b2 * COUT + col) * (HOUT * WOUT) + s] = v;
        } else {
            out[(long)row * COUT + col] = v;
        }
    }
}

// ---------------- maxpool 3x3 stride1 pad1 + ReLU ----------------
template<int C, int H, int W, bool NCHW>
__global__ void maxpool3_relu(const float* __restrict__ in, float* __restrict__ out, int total) {
    int tid = blockIdx.x * blockDim.x + threadIdx.x;
    if (tid >= total) return;
    int b, c, y, x;
    if (NCHW) {
        x = tid % W; int t = tid / W;
        y = t % H;   t /= H;
        c = t % C;   b = t / C;
    } else {
        c = tid % C; int t = tid / C;
        x = t % W;   t /= W;
        y = t % H;   b = t / H;
    }
    float v = NEG_BIG;
    #pragma unroll
    for (int dy = -1; dy <= 1; ++dy) {
        const int yy = y + dy;
        if (yy < 0 || yy >= H) continue;
        #pragma unroll
        for (int dx = -1; dx <= 1; ++dx) {
            const int xx = x + dx;
            if (xx < 0 || xx >= W) continue;
            float s;
            if (NCHW) s = in[(((long)b * C + c) * H + yy) * W + xx];
            else      s = in[(((long)b * H + yy) * W + xx) * C + c];
            v = fmaxf(v, s);
        }
    }
    out[tid] = fmaxf(v, 0.0f);   // ReLU(maxpool(conv))
}

// ---------------- FC: (16 x 61440) x (61440 x 256) via fp32 WMMA, K-split ----------------
__global__ __launch_bounds__(32)
void fc_partial_wmma(const float* __restrict__ A,    // (16, FC_IN)  NCHW-flattened acts
                     const float* __restrict__ Wm,   // fc_w (FC_OUT, FC_IN) row-major
                     float* __restrict__ part)       // (KSPLIT, 16, FC_OUT)
{
    const int lane = threadIdx.x;
    const int lm = lane & 15;
    const int hi = lane >> 4;
    const int nBase = blockIdx.x * 16;
    const int kChunk = FC_IN / KSPLIT;             // 960
    const int k0 = blockIdx.y * kChunk;
    const int n = nBase + lm;

    const float* ap = A  + (long)lm * FC_IN + 2 * hi;   // A[m=lm][k + 2*hi]
    const float* bp = Wm + (long)n  * FC_IN + 2 * hi;   // B[k][n] = fc_w[n][k] (transposed read)

    f32x8 acc = {};
    for (int k = k0; k < k0 + kChunk; k += 4) {
        f32x2 a  = *(const f32x2*)(ap + k);
        f32x2 bf = *(const f32x2*)(bp + k);
        acc = __builtin_amdgcn_wmma_f32_16x16x4_f32(
                  false, a, false, bf, (short)0, acc, false, false);
    }
    float* pout = part + (long)blockIdx.y * 16 * FC_OUT;
    #pragma unroll
    for (int r = 0; r < 8; ++r) {
        const int row = r + 8 * hi;
        pout[(long)row * FC_OUT + nBase + lm] = acc[r];
    }
}

__global__ void fc_reduce(const float* __restrict__ part, const float* __restrict__ fcb,
                          float* __restrict__ out) {
    int tid = blockIdx.x * blockDim.x + threadIdx.x;   // 0 .. 16*FC_OUT-1
    if (tid >= 16 * FC_OUT) return;
    float s = fcb[tid % FC_OUT];
    for (int ks = 0; ks < KSPLIT; ++ks)                // fixed order -> deterministic
        s += part[(long)ks * 16 * FC_OUT + tid];
    out[tid] = fmaxf(s, 0.0f);
}

// ---------------- driver ----------------
extern "C" void kernel_launch(void* const* d_in, const int* in_sizes, int n_in,
                              void* d_out, int out_size, void* d_ws, size_t ws_size,
                              hipStream_t stream) {
    const float* input = (const float*)d_in[0];
    const int*   coords = (const int*)d_in[1];
    const float* w1 = (const float*)d_in[2];  const float* b1 = (const float*)d_in[3];
    const float* w2 = (const float*)d_in[4];  const float* b2 = (const float*)d_in[5];
    const float* w3 = (const float*)d_in[6];  const float* b3 = (const float*)d_in[7];
    const float* fcw = (const float*)d_in[8]; const float* fcb = (const float*)d_in[9];
    const int nActive = in_sizes[1] / 3;

    // workspace carve-out (floats); total ~14.6M floats ~= 59 MB
    float* ws = (float*)d_ws;
    size_t off = 0;
    float* mask = ws + off;  off += (size_t)BATCH * H0 * W0;
    float* wt1  = ws + off;  off += (size_t)9 * C0 * C1;
    float* wt2  = ws + off;  off += (size_t)9 * C1 * C2;
    float* wt3  = ws + off;  off += (size_t)9 * C2 * C3;
    float* a1   = ws + off;  off += (size_t)BATCH * H1 * W1 * C1;
    float* p1   = ws + off;  off += (size_t)BATCH * H1 * W1 * C1;
    float* a2   = ws + off;  off += (size_t)BATCH * H2 * W2 * C2;
    float* p2   = ws + off;  off += (size_t)BATCH * H2 * W2 * C2;
    float* a3   = ws + off;  off += (size_t)BATCH * C3 * H3 * W3;   // NCHW
    float* p3   = ws + off;  off += (size_t)BATCH * C3 * H3 * W3;   // NCHW
    float* fcp  = ws + off;  off += (size_t)KSPLIT * 16 * FC_OUT;

    // 1) sparse mask (avoids materializing the dense masked tensor)
    const int maskN = BATCH * H0 * W0;
    zero_f32<<<(maskN + 255) / 256, 256, 0, stream>>>(mask, maskN);
    scatter_mask<<<(nActive + 255) / 256, 256, 0, stream>>>(coords, mask, nActive);

    // 2) weight repack (tiny)
    repack_weights<C0, C1><<<(9*C0*C1 + 255) / 256, 256, 0, stream>>>(w1, wt1);
    repack_weights<C1, C2><<<(9*C1*C2 + 255) / 256, 256, 0, stream>>>(w2, wt2);
    repack_weights<C2, C3><<<(9*C2*C3 + 255) / 256, 256, 0, stream>>>(w3, wt3);

    // 3) stage 1: conv(masked input) -> pool+relu   (NHWC)
    {
        dim3 g(BATCH * H1 * W1 / 16, C1 / 16);
        conv3x3s2_wmma<C0, C1, H0, W0, H1, W1, true, false>
            <<<g, 32, 0, stream>>>(input, mask, wt1, b1, a1);
        const int t = BATCH * H1 * W1 * C1;
        maxpool3_relu<C1, H1, W1, false><<<(t + 255) / 256, 256, 0, stream>>>(a1, p1, t);
    }
    // 4) stage 2 (NHWC)
    {
        dim3 g(BATCH * H2 * W2 / 16, C2 / 16);
        conv3x3s2_wmma<C1, C2, H1, W1, H2, W2, false, false>
            <<<g, 32, 0, stream>>>(p1, nullptr, wt2, b2, a2);
        const int t = BATCH * H2 * W2 * C2;
        maxpool3_relu<C2, H2, W2, false><<<(t + 255) / 256, 256, 0, stream>>>(a2, p2, t);
    }
    // 5) stage 3: emit NCHW so flatten order matches reference reshape
    {
        dim3 g(BATCH * H3 * W3 / 16, C3 / 16);
        conv3x3s2_wmma<C2, C3, H2, W2, H3, W3, false, true>
            <<<g, 32, 0, stream>>>(p2, nullptr, wt3, b3, a3);
        const int t = BATCH * C3 * H3 * W3;
        maxpool3_relu<C3, H3, W3, true><<<(t + 255) / 256, 256, 0, stream>>>(a3, p3, t);
    }
    // 6) FC + ReLU (WMMA, deterministic K-split reduction)
    {
        dim3 g(FC_OUT / 16, KSPLIT);
        fc_partial_wmma<<<g, 32, 0, stream>>>(p3, fcw, fcp);
        fc_reduce<<<(16 * FC_OUT + 255) / 256, 256, 0, stream>>>(fcp, fcb, (float*)d_out);
    }
}